// GraphAutoencoder_77043123356189
// MI455X (gfx1250) — compile-verified
//
#include <hip/hip_runtime.h>
#include <hip/hip_bf16.h>

typedef __attribute__((ext_vector_type(2))) float v2f;
typedef __attribute__((ext_vector_type(8))) float v8f;

#define NN    10000
#define IN_C  256
#define HID   128
#define LAT   64

// ---------------------------------------------------------------------------
// Degree / normalization kernels:  deg = 1 (self loop) + #incoming edges,
// then dinv = rsqrt(max(deg,1)).
// ---------------------------------------------------------------------------
__global__ void deg_init(float* __restrict__ deg, int n) {
    int i = blockIdx.x * blockDim.x + threadIdx.x;
    if (i < n) deg[i] = 1.0f;   // self-loop contribution
}

__global__ void deg_edges(const int* __restrict__ ei, int E, float* __restrict__ deg) {
    int e = blockIdx.x * blockDim.x + threadIdx.x;
    if (e < E) {
        int dst = ei[E + e];
        atomicAdd(&deg[dst], 1.0f);
    }
}

__global__ void deg_finish(float* __restrict__ deg, int n) {
    int i = blockIdx.x * blockDim.x + threadIdx.x;
    if (i < n) deg[i] = rsqrtf(fmaxf(deg[i], 1.0f));   // deg buffer becomes dinv
}

// ---------------------------------------------------------------------------
// Dense GEMM  C[N x M] = A[N x K] * B[K x M]  via V_WMMA_F32_16X16X4_F32.
// One wave (32 lanes) computes one 16x16 tile of C; K stepped by 4.
// Lane layout per CDNA5 ISA 7.12.2 (32-bit A 16x4 / B 4x16 / C,D 16x16):
//   A: lanes 0-15 rows, VGPR0 = K0 (half 1: K2), VGPR1 = K1 (half 1: K3)
//   B: lanes 0-15 cols, VGPR0 = K0-row (half 1: K2), VGPR1 = K1 (K3)
//   D: VGPR r -> M = r (lanes 0-15) / r+8 (lanes 16-31), N = lane&15
// ---------------------------------------------------------------------------
__global__ __launch_bounds__(32) void gemm_wmma_f32(
    const float* __restrict__ A, const float* __restrict__ B,
    float* __restrict__ C, int K, int M)
{
    const int lane    = threadIdx.x;
    const int half    = lane >> 4;       // 0: K {0,1};  1: K {2,3}
    const int l15     = lane & 15;
    const int rowBase = blockIdx.y << 4;
    const int colBase = blockIdx.x << 4;
    const int row     = rowBase + l15;
    const int col     = colBase + l15;

    v8f acc = {};
    for (int kk = 0; kk < K; kk += 4) {
        const int k0 = kk + half * 2;
        v2f a, b;
        a.x = A[row * K + k0];
        a.y = A[row * K + k0 + 1];
        b.x = B[(k0)     * M + col];
        b.y = B[(k0 + 1) * M + col];
        acc = __builtin_amdgcn_wmma_f32_16x16x4_f32(
                  false, a, false, b, (short)0, acc, false, false);
    }
#pragma unroll
    for (int r = 0; r < 8; ++r) {
        int orow = rowBase + r + half * 8;
        C[orow * M + col] = acc[r];
    }
}

// ---------------------------------------------------------------------------
// adj = S * S^T  (N x N, K = LAT = 64), fp32 WMMA, 8 waves / block,
// each wave one 16x16 tile.  B[k][n] = S[n][k] (gather transposed).
// ---------------------------------------------------------------------------
__global__ __launch_bounds__(256) void adj_gemm_wmma(
    const float* __restrict__ S, float* __restrict__ C)
{
    const int lane = threadIdx.x & 31;
    const int wave = threadIdx.x >> 5;
    const int nT   = NN / 16;                    // 625 tiles per dim
    const int ct   = blockIdx.x * 8 + wave;      // column tile
    const int rt   = blockIdx.y;                 // row tile
    if (ct >= nT) return;                        // wave-uniform exit (EXEC all-1 for WMMA)

    const int half    = lane >> 4;
    const int l15     = lane & 15;
    const int rowBase = rt << 4;
    const int colBase = ct << 4;
    const int row     = rowBase + l15;
    const int col     = colBase + l15;

    v8f acc = {};
#pragma unroll
    for (int kk = 0; kk < LAT; kk += 4) {
        const int k0 = kk + half * 2;
        v2f a, b;
        a.x = S[row * LAT + k0];
        a.y = S[row * LAT + k0 + 1];
        b.x = S[col * LAT + k0];       // B[k][n] = S[n][k]
        b.y = S[col * LAT + k0 + 1];
        acc = __builtin_amdgcn_wmma_f32_16x16x4_f32(
                  false, a, false, b, (short)0, acc, false, false);
    }
#pragma unroll
    for (int r = 0; r < 8; ++r) {
        int orow = rowBase + r + half * 8;
        C[(size_t)orow * NN + col] = acc[r];
    }
}

// ---------------------------------------------------------------------------
// Aggregation:  out = relu( b + sum_{edges+self} dinv[s]*dinv[d] * hW[s] )
// split into: self-loop init  ->  per-edge atomic scatter  ->  bias+relu
// ---------------------------------------------------------------------------
__global__ void self_init(const float* __restrict__ tmp, const float* __restrict__ dinv,
                          float* __restrict__ out, int M)
{
    int idx = blockIdx.x * blockDim.x + threadIdx.x;
    if (idx < NN * M) {
        int i = idx / M;
        float d = dinv[i];
        out[idx] = d * d * tmp[idx];
    }
}

__global__ __launch_bounds__(256) void edge_agg(
    const int* __restrict__ ei, int E,
    const float* __restrict__ tmp, const float* __restrict__ dinv,
    float* __restrict__ out, int M)
{
    const int wave = threadIdx.x >> 5;
    const int lane = threadIdx.x & 31;
    const int e    = blockIdx.x * 8 + wave;      // one wave per edge
    if (e >= E) return;
    const int src = ei[e];
    const int dst = ei[E + e];
    const float w = dinv[src] * dinv[dst];
    const float* __restrict__ srow = tmp + (size_t)src * M;
    float* __restrict__       drow = out + (size_t)dst * M;
    for (int m = lane; m < M; m += 32)
        atomicAdd(&drow[m], w * srow[m]);
}

__global__ void bias_relu(float* __restrict__ out, const float* __restrict__ b, int M)
{
    int idx = blockIdx.x * blockDim.x + threadIdx.x;
    if (idx < NN * M)
        out[idx] = fmaxf(out[idx] + b[idx % M], 0.0f);
}

// ---------------------------------------------------------------------------
// Host-side orchestration (graph-capture safe: only launches on `stream`).
// ---------------------------------------------------------------------------
static void gcn_layer(const float* X, const float* W, const float* bias,
                      float* out, float* tmp, const float* dinv,
                      const int* ei, int E, int K, int M, hipStream_t stream)
{
    dim3 g(M / 16, NN / 16);
    gemm_wmma_f32<<<g, 32, 0, stream>>>(X, W, tmp, K, M);
    int tot = NN * M;
    self_init<<<(tot + 255) / 256, 256, 0, stream>>>(tmp, dinv, out, M);
    edge_agg<<<(E + 7) / 8, 256, 0, stream>>>(ei, E, tmp, dinv, out, M);
    bias_relu<<<(tot + 255) / 256, 256, 0, stream>>>(out, bias, M);
}

extern "C" void kernel_launch(void* const* d_in, const int* in_sizes, int n_in,
                              void* d_out, int out_size, void* d_ws, size_t ws_size,
                              hipStream_t stream)
{
    const float* x    = (const float*)d_in[0];
    const int*   ei   = (const int*)d_in[1];
    const int    E    = in_sizes[1] / 2;
    const float* W_e1 = (const float*)d_in[2];  const float* b_e1 = (const float*)d_in[3];
    const float* W_e2 = (const float*)d_in[4];  const float* b_e2 = (const float*)d_in[5];
    const float* W_a1 = (const float*)d_in[6];  const float* b_a1 = (const float*)d_in[7];
    const float* W_a2 = (const float*)d_in[8];  const float* b_a2 = (const float*)d_in[9];
    const float* W_s  = (const float*)d_in[10]; const float* b_s  = (const float*)d_in[11];

    // workspace layout (floats): dinv | tmp(N*256) | h/a(N*128) | z(N*64) | s(N*64)
    float* ws   = (float*)d_ws;
    float* dinv = ws;
    float* tmp  = dinv + NN;
    float* h    = tmp  + (size_t)NN * IN_C;
    float* z    = h    + (size_t)NN * HID;
    float* s    = z    + (size_t)NN * LAT;

    float* x_recon = (float*)d_out;                    // N x 256
    float* adj     = x_recon + (size_t)NN * IN_C;      // N x N

    // normalization coefficients
    deg_init  <<<(NN + 255) / 256, 256, 0, stream>>>(dinv, NN);
    deg_edges <<<(E  + 255) / 256, 256, 0, stream>>>(ei, E, dinv);
    deg_finish<<<(NN + 255) / 256, 256, 0, stream>>>(dinv, NN);

    // encoder
    gcn_layer(x, W_e1, b_e1, h, tmp, dinv, ei, E, IN_C, HID, stream);   // h
    gcn_layer(h, W_e2, b_e2, z, tmp, dinv, ei, E, HID,  LAT, stream);   // z
    // attribute decoder (reuse h buffer for 'a')
    gcn_layer(z, W_a1, b_a1, h, tmp, dinv, ei, E, LAT,  HID, stream);   // a
    gcn_layer(h, W_a2, b_a2, x_recon, tmp, dinv, ei, E, HID, IN_C, stream);
    // structure decoder
    gcn_layer(z, W_s, b_s, s, tmp, dinv, ei, E, LAT, LAT, stream);      // s

    // adj_recon = s @ s^T  (store-bandwidth bound: 400 MB out)
    dim3 ga((NN / 16 + 7) / 8, NN / 16);
    adj_gemm_wmma<<<ga, 256, 0, stream>>>(s, adj);
}